// PLNet_60911226191951
// MI455X (gfx1250) — compile-verified
//
#include <hip/hip_runtime.h>
#include <hip/hip_bf16.h>

typedef __attribute__((ext_vector_type(2))) float v2f;
typedef __attribute__((ext_vector_type(8))) float v8f;
typedef __attribute__((ext_vector_type(4))) unsigned int u32x4;
typedef __attribute__((ext_vector_type(8))) int i32x8;
typedef __attribute__((ext_vector_type(4))) int i32x4;

#define SP 196          // 14x14 spatial
#define NK 20
#define LCH 28          // channels 23..50 staged per side (Lx: 0..13, Ly: 14..27)

// ---- Tensor Data Mover: 1-D slab load, D# per CDNA5 ISA ----
// Group0: count=1 | lds_addr[63:32] | global_addr[120:64] | type[127:126]=2
// Group1: data_size[17:16]=2 (4B) | tensor_dim0[79:48] | tensor_dim1[111:80]=1
//         | tile_dim0[127:112] | tile_dim1[143:128]=1 | tensor_dim0_stride[207:160]
__device__ __forceinline__ void tdm_load_slab(const float* gsrc, unsigned lds_off,
                                              unsigned nelem) {
    unsigned long long ga = (unsigned long long)(uintptr_t)gsrc;
    u32x4 g0;
    g0.x = 1u;                                               // count=1 (valid user D#)
    g0.y = lds_off;                                          // lds_addr (bytes)
    g0.z = (unsigned)(ga & 0xFFFFFFFFu);                     // global_addr[31:0]
    g0.w = (unsigned)((ga >> 32) & 0x01FFFFFFu) | (2u << 30);// global_addr[56:32], type=2

    i32x8 g1;
    g1[0] = 0x20000;                                         // data_size=2 -> 4 bytes
    g1[1] = (int)((nelem & 0xFFFFu) << 16);                  // tensor_dim0[63:48]
    g1[2] = (int)((nelem >> 16) & 0xFFFFu) | (1 << 16);      // tensor_dim0[79:64], tensor_dim1=1
    g1[3] = (int)(nelem << 16);                              // tile_dim0[127:112]
    g1[4] = 1;                                               // tile_dim1=1
    g1[5] = (int)nelem;                                      // tensor_dim0_stride low
    g1[6] = 0;
    g1[7] = 0;

    i32x4 z4 = {0, 0, 0, 0};                                 // groups 2/3 unused (<=2D)
    i32x8 z8 = {0, 0, 0, 0, 0, 0, 0, 0};
    // amdgpu-toolchain (clang-23) 6-arg form: (g0, g1, g2, g3, g4, cpol)
    __builtin_amdgcn_tensor_load_to_lds(g0, g1, z4, z4, z8, 0);
}

// One block per (n, pairing pc, k). 256 threads = 8 wave32s.
// Per (i,j) tile: one v_wmma_f32_16x16x4_f32 rank-1 outer product in native
// C/D layout, elementwise F2 from LDS, masked row stores.
__global__ __launch_bounds__(256) void plnet_poss_kernel(
    const float* __restrict__ in,   // (N, 204, 14, 14)
    float* __restrict__ out,        // (p1..p4) x (N, 20, 14,14, 14,14)
    int N)
{
    __shared__ float s_cor[SP];           // 0.5 * cor0 * cor(1+k)
    __shared__ float s_cen[SP];           //       cen0 * cen(1+k)
    __shared__ float corL[LCH * SP];      // cor ch 23..50: [0..13]=Lx(u), [14..27]=Ly(v)
    __shared__ float cenL[LCH * SP];      // cen ch 23..50: [0..13]=Lx(i), [14..27]=Ly(j)

    const int tid = threadIdx.x;
    const int bid = blockIdx.x;
    const int n   = bid / (4 * NK);
    const int rem = bid % (4 * NK);
    const int pc  = rem / NK;
    const int k   = rem % NK;
    const int cor_idx = pc & 1;
    const int cen_idx = pc >> 1;

    const float* corp = in + (size_t)(n * 204 + cor_idx * 51) * SP;
    const float* cenp = in + (size_t)(n * 204 + (2 + cen_idx) * 51) * SP;

    const int wave = tid >> 5;            // 8 waves (wave32)
    const int lane = tid & 31;

    // ---- Async-tensor staging of the two contiguous 28-channel slabs ----
    if (wave == 0) {
        tdm_load_slab(corp + 23 * SP,
                      (unsigned)(unsigned long long)(uintptr_t)(void*)corL,
                      LCH * SP);
        tdm_load_slab(cenp + 23 * SP,
                      (unsigned)(unsigned long long)(uintptr_t)(void*)cenL,
                      LCH * SP);
        __builtin_amdgcn_s_wait_tensorcnt(0);
    }
    // Small product images via plain loads (need VALU anyway).
    if (tid < SP) {
        s_cor[tid] = 0.5f * corp[tid] * corp[(1 + k) * SP + tid];
        s_cen[tid] =        cenp[tid] * cenp[(1 + k) * SP + tid];
    }
    __syncthreads();

    const int hi  = (lane >> 4) & 1;      // C/D layout: u = r + 8*hi
    const int v   = lane & 15;            // C/D layout: v = lane % 16
    const int vc  = (v < 14) ? v : 13;    // clamped for LDS reads
    const bool vok = (v < 14);
    const int lc  = (lane < 14) ? lane : 13;  // clamped channel for A/B gather

    const size_t per_p = (size_t)N * NK * SP * SP;
    float* outp = out + (size_t)pc * per_p + ((size_t)(n * NK + k)) * SP * SP;

    int i = wave / 14, j = wave % 14;     // wave 0..7 -> i=0, j=wave
    for (int t = wave; t < SP; t += 8) {
        // Branchless A/B: unconditional clamped LDS load + value select.
        const float av = s_cor[t] * corL[lc * SP + t];        // a[u] = s * cor[23+u](i,j)
        const float bv = corL[(14 + lc) * SP + t];            // b[v] = cor[37+v](i,j)
        v2f A, B;
        A.x = (lane < 14) ? av : 0.0f;  A.y = 0.0f;           // K=0 column only
        B.x = (lane < 14) ? bv : 0.0f;  B.y = 0.0f;           // K=0 row only

        v8f C = {};
        v8f D = __builtin_amdgcn_wmma_f32_16x16x4_f32(
            false, A, false, B, (short)0, C, false, false);   // D[u,v] = a[u]*b[v]

        float* tout = outp + (size_t)t * SP;
        const float* lx = &cenL[i * SP];                      // cen[23+i](u,v)
        const float* ly = &cenL[(14 + j) * SP];               // cen[37+j](u,v)

        // Rows r=0..5: u = r+8*hi in [0..13] for both half-waves.
#pragma unroll
        for (int r = 0; r < 6; ++r) {
            const int uv = (r + (hi ? 8 : 0)) * 14 + vc;
            const float val = D[r] * s_cen[uv] * lx[uv] * ly[uv];
            if (vok) tout[uv] = val;
        }
        // Rows r=6,7: valid only for the low half-wave (u = r).
#pragma unroll
        for (int r = 6; r < 8; ++r) {
            const int uv = r * 14 + vc;
            const float val = D[r] * s_cen[uv] * lx[uv] * ly[uv];
            if (vok && !hi) tout[uv] = val;
        }

        // Incremental (i,j) update for t += 8 (no div/mod in the loop).
        j += 8;
        if (j >= 14) { j -= 14; ++i; }
    }
}

extern "C" void kernel_launch(void* const* d_in, const int* in_sizes, int n_in,
                              void* d_out, int out_size, void* d_ws, size_t ws_size,
                              hipStream_t stream) {
    const float* in = (const float*)d_in[0];
    float* out = (float*)d_out;
    const int N = in_sizes[0] / (204 * 14 * 14);   // 32

    const int blocks = N * 4 * NK;                 // 2560 blocks, 256 threads each
    plnet_poss_kernel<<<blocks, 256, 0, stream>>>(in, out, N);
}